// GAT_13099650253145
// MI455X (gfx1250) — compile-verified
//
#include <hip/hip_runtime.h>
#include <hip/hip_bf16.h>

// Problem constants (match reference)
#define NNODES 50000
#define NEDGES 800000
#define ETOT   (NEDGES + NNODES)   // edges + self loops
#define NHEADS 4
#define CFEAT  64
#define FDIM   256                 // H*C (both layers, and C_IN)
#define NEG_SLOPE 0.2f

typedef __attribute__((ext_vector_type(2))) float v2f;
typedef __attribute__((ext_vector_type(8))) float v8f;

// ---------------------------------------------------------------------------
// GEMM: P[n_rows x 256] = X[n_rows x 256] * W[256 x 256], fp32 WMMA 16x16x4.
// One wave per 16x16 output tile; K loop of 64 chained v_wmma_f32_16x16x4_f32.
// A layout (ISA 7.12.2): lanes 0-15 -> M=0..15 {K=k,k+1}; lanes 16-31 -> {K=k+2,k+3}
// B layout: VGPR0 = row k (lanes0-15) / row k+2 (lanes16-31); VGPR1 = k+1 / k+3
// C/D layout: VGPR r -> M=r (lanes0-15), M=8+r (lanes16-31); N = lane&15
// ---------------------------------------------------------------------------
__global__ void gat_gemm_f32_wmma(const float* __restrict__ X,
                                  const float* __restrict__ W,
                                  float* __restrict__ P,
                                  int n_rows) {
    const int lane  = threadIdx.x & 31;
    const int wave  = threadIdx.x >> 5;
    const int wpb   = blockDim.x >> 5;
    const int tile  = blockIdx.x * wpb + wave;
    const int tilesN = FDIM / 16;                  // 16
    const int tiles  = (n_rows / 16) * tilesN;
    if (tile >= tiles) return;

    const int tm = (tile / tilesN) * 16;
    const int tn = (tile % tilesN) * 16;
    const int half = lane >> 4;                    // 0: K pair 0/1, 1: K pair 2/3
    const int l    = lane & 15;

    const float* xrow = X + (size_t)(tm + l) * FDIM;
    v8f acc = {};

    #pragma unroll 8
    for (int k = 0; k < FDIM; k += 4) {
        v2f a;
        {   // contiguous 8B load per lane
            const float2 av = *(const float2*)(xrow + k + half * 2);
            a.x = av.x; a.y = av.y;
        }
        v2f b;
        b.x = W[(size_t)(k + half * 2)     * FDIM + tn + l];
        b.y = W[(size_t)(k + half * 2 + 1) * FDIM + tn + l];
        acc = __builtin_amdgcn_wmma_f32_16x16x4_f32(
                  /*neg_a=*/false, a, /*neg_b=*/false, b,
                  /*c_mod=*/(short)0, acc, /*reuse_a=*/false, /*reuse_b=*/false);
    }

    float* out = P + (size_t)(tm + half * 8) * FDIM + tn + l;
    #pragma unroll
    for (int r = 0; r < 8; ++r)
        out[(size_t)r * FDIM] = acc[r];
}

// ---------------------------------------------------------------------------
// Per-node attention logits: a_src[n,h] = sum_c P[n,h,c]*att_src[h,c] (same dst)
// ---------------------------------------------------------------------------
__global__ void gat_attn_scores(const float* __restrict__ P,
                                const float* __restrict__ att_src,
                                const float* __restrict__ att_dst,
                                float* __restrict__ a_src,
                                float* __restrict__ a_dst,
                                int n_rows) {
    int idx = blockIdx.x * blockDim.x + threadIdx.x;      // n*H + h
    if (idx >= n_rows * NHEADS) return;
    const int h = idx & (NHEADS - 1);
    const float* p  = P + (size_t)(idx >> 2) * FDIM + h * CFEAT;
    const float* as = att_src + h * CFEAT;
    const float* ad = att_dst + h * CFEAT;
    float ss = 0.f, sd = 0.f;
    #pragma unroll 8
    for (int c = 0; c < CFEAT; ++c) {
        float v = p[c];
        ss += v * as[c];
        sd += v * ad[c];
    }
    a_src[idx] = ss;
    a_dst[idx] = sd;
}

// Ordered-uint encoding so atomicMax(u32) implements float max (finite values)
__device__ __forceinline__ unsigned enc_f32(float f) {
    unsigned u = __float_as_uint(f);
    return (u & 0x80000000u) ? ~u : (u | 0x80000000u);
}
__device__ __forceinline__ float dec_f32(unsigned u) {
    unsigned v = (u & 0x80000000u) ? (u & 0x7FFFFFFFu) : ~u;
    return __uint_as_float(v);
}

__global__ void gat_zero_f32(float* __restrict__ p, size_t n) {
    size_t i = (size_t)blockIdx.x * blockDim.x + threadIdx.x;
    if (i < n) p[i] = 0.f;
}
__global__ void gat_zero_u32(unsigned* __restrict__ p, size_t n) {
    size_t i = (size_t)blockIdx.x * blockDim.x + threadIdx.x;
    if (i < n) p[i] = 0u;      // 0 encodes "below any finite float"
}

// ---------------------------------------------------------------------------
// Edge pass 1: alpha = leaky_relu(a_src[s]+a_dst[d]); segment max via atomicMax
// Edges e >= NEDGES are implicit self loops (s = d = e - NEDGES).
// ---------------------------------------------------------------------------
__global__ void gat_edge_alpha(const int* __restrict__ src,
                               const int* __restrict__ dst,
                               const float* __restrict__ a_src,
                               const float* __restrict__ a_dst,
                               float* __restrict__ alpha_e,
                               unsigned* __restrict__ m_enc) {
    int e = blockIdx.x * blockDim.x + threadIdx.x;
    if (e >= ETOT) return;
    const int s = (e < NEDGES) ? src[e] : (e - NEDGES);
    const int d = (e < NEDGES) ? dst[e] : (e - NEDGES);
    #pragma unroll
    for (int h = 0; h < NHEADS; ++h) {
        float a = a_src[s * NHEADS + h] + a_dst[d * NHEADS + h];
        a = (a >= 0.f) ? a : NEG_SLOPE * a;
        alpha_e[(size_t)e * NHEADS + h] = a;
        atomicMax(&m_enc[d * NHEADS + h], enc_f32(a));
    }
}

// Edge pass 2: e = exp(alpha - m[dst]); denom[dst] += e  (overwrite alpha_e)
__global__ void gat_edge_expsum(const int* __restrict__ dst,
                                float* __restrict__ alpha_e,
                                const unsigned* __restrict__ m_enc,
                                float* __restrict__ denom) {
    int e = blockIdx.x * blockDim.x + threadIdx.x;
    if (e >= ETOT) return;
    const int d = (e < NEDGES) ? dst[e] : (e - NEDGES);
    #pragma unroll
    for (int h = 0; h < NHEADS; ++h) {
        float m  = dec_f32(m_enc[d * NHEADS + h]);
        float ex = expf(alpha_e[(size_t)e * NHEADS + h] - m);
        alpha_e[(size_t)e * NHEADS + h] = ex;
        atomicAdd(&denom[d * NHEADS + h], ex);
    }
}

// Edge pass 3: agg[dst, f] += (e/denom) * P[src, f]. One 256-thread block/edge:
// coalesced 1KB read of P[src] (L2 resident), 256 float atomics to agg[dst].
__global__ void gat_edge_aggregate(const int* __restrict__ src,
                                   const int* __restrict__ dst,
                                   const float* __restrict__ alpha_e,
                                   const float* __restrict__ denom,
                                   const float* __restrict__ P,
                                   float* __restrict__ agg) {
    const int e = blockIdx.x;
    const int f = threadIdx.x;             // 0..255
    const int h = f >> 6;                  // feature f belongs to head f/64
    const int s = (e < NEDGES) ? src[e] : (e - NEDGES);
    const int d = (e < NEDGES) ? dst[e] : (e - NEDGES);
    float den = denom[d * NHEADS + h];
    den = (den > 1e-16f) ? den : 1e-16f;
    const float w = alpha_e[(size_t)e * NHEADS + h] / den;
    atomicAdd(&agg[(size_t)d * FDIM + f], w * P[(size_t)s * FDIM + f]);
}

// out = agg + bias, optional ReLU (in place is fine: elementwise)
__global__ void gat_epilogue(const float* __restrict__ agg,
                             const float* __restrict__ bias,
                             float* __restrict__ out,
                             int n_rows, int do_relu) {
    size_t idx = (size_t)blockIdx.x * blockDim.x + threadIdx.x;
    if (idx >= (size_t)n_rows * FDIM) return;
    const int f = (int)(idx & (FDIM - 1));
    float v = agg[idx] + bias[f];
    if (do_relu) v = fmaxf(v, 0.f);
    out[idx] = v;
}

// ---------------------------------------------------------------------------
extern "C" void kernel_launch(void* const* d_in, const int* in_sizes, int n_in,
                              void* d_out, int out_size, void* d_ws, size_t ws_size,
                              hipStream_t stream) {
    const float* x   = (const float*)d_in[0];
    const int*   ei  = (const int*)  d_in[1];
    const int*   src = ei;
    const int*   dst = ei + NEDGES;
    const float* W1  = (const float*)d_in[2];
    const float* as1 = (const float*)d_in[3];
    const float* ad1 = (const float*)d_in[4];
    const float* b1  = (const float*)d_in[5];
    const float* W2  = (const float*)d_in[6];
    const float* as2 = (const float*)d_in[7];
    const float* ad2 = (const float*)d_in[8];
    const float* b2  = (const float*)d_in[9];

    char* w = (char*)d_ws;
    float*    bufP   = (float*)w;    w += (size_t)NNODES * FDIM * 4;     // 51.2 MB
    float*    bufAgg = (float*)w;    w += (size_t)NNODES * FDIM * 4;     // 51.2 MB
    float*    aS     = (float*)w;    w += (size_t)NNODES * NHEADS * 4;
    float*    aD     = (float*)w;    w += (size_t)NNODES * NHEADS * 4;
    unsigned* mE     = (unsigned*)w; w += (size_t)NNODES * NHEADS * 4;
    float*    den    = (float*)w;    w += (size_t)NNODES * NHEADS * 4;
    float*    alphaE = (float*)w;    w += (size_t)ETOT   * NHEADS * 4;   // 13.6 MB

    const int gemmTiles  = (NNODES / 16) * (FDIM / 16);                  // 50000 waves
    const int gemmBlocks = (gemmTiles + 3) / 4;                          // 4 waves/block
    const int nhTotal    = NNODES * NHEADS;
    const int nhBlocks   = (nhTotal + 255) / 256;
    const int eBlocks    = (ETOT + 255) / 256;
    const size_t nf      = (size_t)NNODES * FDIM;
    const int nfBlocks   = (int)((nf + 255) / 256);

    float* outF = (float*)d_out;

    // ----------------- Layer 1 -----------------
    gat_gemm_f32_wmma<<<gemmBlocks, 128, 0, stream>>>(x, W1, bufP, NNODES);
    gat_attn_scores  <<<nhBlocks, 256, 0, stream>>>(bufP, as1, ad1, aS, aD, NNODES);
    gat_zero_u32     <<<nhBlocks, 256, 0, stream>>>(mE,  (size_t)nhTotal);
    gat_zero_f32     <<<nhBlocks, 256, 0, stream>>>(den, (size_t)nhTotal);
    gat_zero_f32     <<<nfBlocks, 256, 0, stream>>>(bufAgg, nf);
    gat_edge_alpha   <<<eBlocks, 256, 0, stream>>>(src, dst, aS, aD, alphaE, mE);
    gat_edge_expsum  <<<eBlocks, 256, 0, stream>>>(dst, alphaE, mE, den);
    gat_edge_aggregate<<<ETOT, 256, 0, stream>>>(src, dst, alphaE, den, bufP, bufAgg);
    gat_epilogue     <<<nfBlocks, 256, 0, stream>>>(bufAgg, b1, bufAgg, NNODES, 1);
    // bufAgg now holds h1 = relu(layer1 out)

    // ----------------- Layer 2 -----------------
    gat_gemm_f32_wmma<<<gemmBlocks, 128, 0, stream>>>(bufAgg, W2, bufP, NNODES);
    gat_attn_scores  <<<nhBlocks, 256, 0, stream>>>(bufP, as2, ad2, aS, aD, NNODES);
    gat_zero_u32     <<<nhBlocks, 256, 0, stream>>>(mE,  (size_t)nhTotal);
    gat_zero_f32     <<<nhBlocks, 256, 0, stream>>>(den, (size_t)nhTotal);
    gat_zero_f32     <<<nfBlocks, 256, 0, stream>>>(outF, nf);
    gat_edge_alpha   <<<eBlocks, 256, 0, stream>>>(src, dst, aS, aD, alphaE, mE);
    gat_edge_expsum  <<<eBlocks, 256, 0, stream>>>(dst, alphaE, mE, den);
    gat_edge_aggregate<<<ETOT, 256, 0, stream>>>(src, dst, alphaE, den, bufP, outF);
    gat_epilogue     <<<nfBlocks, 256, 0, stream>>>(outF, b2, outF, NNODES, 0);
}